// QuantizerEMA_10548439679061
// MI455X (gfx1250) — compile-verified
//
#include <hip/hip_runtime.h>
#include <hip/hip_bf16.h>

typedef __attribute__((ext_vector_type(2))) float v2f;
typedef __attribute__((ext_vector_type(8))) float v8f;
typedef __attribute__((ext_vector_type(2))) int   v2i;

#define DECAY     0.99f
#define ONE_MINUS 0.01f
#define COMMIT    0.25f
#define EPSV      1e-5f

#define N_TOT   65536   // 64*32*32
#define D_DIM   64
#define K_CODES 1024
#define HWSZ    1024    // 32*32

// d_out float offsets
#define OUT_Q    0
#define OUT_IDX  4194304
#define OUT_LOSS 4259840
#define OUT_EMB  4259841
#define OUT_EMA  4325377   // accumulates dw, finalized to new_ema_embed
#define OUT_CS   4390913   // accumulates counts, finalized to new_cluster_size

// ws float offsets: [0,1024) = enorm, [1024] = n accumulator
#define WS_ENORM 0
#define WS_NACC  1024

#if __has_builtin(__builtin_amdgcn_global_load_async_to_lds_b64)
#define HAS_ASYNC 1
typedef __attribute__((address_space(1))) v2i* gptr_v2i;
typedef __attribute__((address_space(3))) v2i* lptr_v2i;
#else
#define HAS_ASYNC 0
#endif

// ---------------- kernel 1: init accumulators + embedding norms ----------------
__global__ void vq_init_kernel(const float* __restrict__ emb,
                               float* __restrict__ out, float* __restrict__ ws) {
    int tid = blockIdx.x * blockDim.x + threadIdx.x;   // 65536 threads
    out[OUT_EMA + tid] = 0.0f;                         // dw accumulator
    if (tid < K_CODES) {
        out[OUT_CS + tid] = 0.0f;                      // counts accumulator
        const float* row = emb + tid * D_DIM;
        float s = 0.0f;
        #pragma unroll 8
        for (int d = 0; d < D_DIM; ++d) { float v = row[d]; s += v * v; }
        ws[WS_ENORM + tid] = s;
    }
    if (tid == 0) { out[OUT_LOSS] = 0.0f; ws[WS_NACC] = 0.0f; }
}

// ---------------- kernel 2: WMMA distance + argmin ----------------
// Block = 8 waves * 16 rows = 128 rows. All waves sweep the same 64 code tiles;
// wave 0 stages each 16x64 embedding tile into LDS (double-buffered, fragment
// order [kc][lane] -> conflict-free ds_load_b64) via async global->LDS copies
// overlapped with the WMMA work on the previous tile.
__global__ void __launch_bounds__(256)
vq_argmin_wmma_kernel(const float* __restrict__ z,
                      const float* __restrict__ emb,
                      const float* __restrict__ ws,
                      float* __restrict__ out) {
    __shared__ float bstage[2][16 * 64];   // 2 x 4KB tiles, [kc][lane][2] floats

    const int lane = threadIdx.x & 31;
    const int half = lane >> 4;        // 0: K={0,1}/rows 0-7 ; 1: K={2,3}/rows 8-15
    const int l16  = lane & 15;
    const int waveIG = (blockIdx.x * blockDim.x + threadIdx.x) >> 5;
    const int rowbase = waveIG * 16;

    // A fragments: 16 K-chunks of 4; this lane supplies 2 consecutive K values
    v2f afrag[16];
    const float* zrow = z + (rowbase + l16) * D_DIM + half * 2;
    #pragma unroll
    for (int kc = 0; kc < 16; ++kc)
        afrag[kc] = *(const v2f*)(zrow + kc * 4);

    float bestd[8];
    int   besti[8];
    #pragma unroll
    for (int m = 0; m < 8; ++m) { bestd[m] = 3.4e38f; besti[m] = 0; }

    // ---- tile staging (wave 0 only; wave-uniform branch keeps EXEC full) ----
    auto stage = [&](int tb, int buf) {
        if (threadIdx.x < 32) {
            const int sl16 = threadIdx.x & 15;
            const int shalf = threadIdx.x >> 4;
            const float* gsrc = emb + (tb * 16 + sl16) * D_DIM + shalf * 2;
            float* ldst = &bstage[buf][threadIdx.x * 2];
#if HAS_ASYNC
            #pragma unroll
            for (int kc = 0; kc < 16; ++kc)
                __builtin_amdgcn_global_load_async_to_lds_b64(
                    (gptr_v2i)(gsrc + kc * 4),
                    (lptr_v2i)(ldst + kc * 64),
                    0, 0);
#else
            #pragma unroll
            for (int kc = 0; kc < 16; ++kc)
                *(v2f*)(ldst + kc * 64) = *(const v2f*)(gsrc + kc * 4);
#endif
        }
    };
    auto tile_sync = [&]() {
#if HAS_ASYNC
        asm volatile("s_wait_asynccnt 0x0" ::: "memory");
#endif
        __syncthreads();
    };

    stage(0, 0);
    tile_sync();

    for (int tb = 0; tb < K_CODES / 16; ++tb) {
        const int cur = tb & 1;
        if (tb + 1 < K_CODES / 16) stage(tb + 1, cur ^ 1);

        const float* bbuf = &bstage[cur][lane * 2];
        // two independent accumulator chains so dependent-WMMA latency overlaps
        v8f c0 = {}, c1 = {};
        #pragma unroll
        for (int kc = 0; kc < 8; ++kc) {
            v2f b0 = *(const v2f*)(bbuf + kc * 64);
            v2f b1 = *(const v2f*)(bbuf + (kc + 8) * 64);
            c0 = __builtin_amdgcn_wmma_f32_16x16x4_f32(
                     false, afrag[kc], false, b0, (short)0, c0, false, false);
            c1 = __builtin_amdgcn_wmma_f32_16x16x4_f32(
                     false, afrag[kc + 8], false, b1, (short)0, c1, false, false);
        }

        const int code = tb * 16 + l16;                 // this lane's column/code
        const float en = ws[WS_ENORM + code];
        #pragma unroll
        for (int m = 0; m < 8; ++m) {
            float s = en - 2.0f * (c0[m] + c1[m]);
            if (s < bestd[m]) { bestd[m] = s; besti[m] = code; }
        }
        tile_sync();   // copy of tile tb+1 done + all waves finished reading
    }

    // reduce argmin across the 16 lanes of each half (xor 1,2,4,8 stays in-group)
    #pragma unroll
    for (int off = 1; off < 16; off <<= 1) {
        #pragma unroll
        for (int m = 0; m < 8; ++m) {
            float od = __shfl_xor(bestd[m], off, 32);
            int   oi = __shfl_xor(besti[m], off, 32);
            if (od < bestd[m] || (od == bestd[m] && oi < besti[m])) {
                bestd[m] = od; besti[m] = oi;
            }
        }
    }

    // lane 0 -> rows 0..7, lane 16 -> rows 8..15
    if (l16 == 0) {
        #pragma unroll
        for (int m = 0; m < 8; ++m)
            out[OUT_IDX + rowbase + half * 8 + m] = (float)besti[m];
    }
}

// ---------------- kernel 3: gather, transpose-store, scatter EMA stats ----------------
__global__ void vq_scatter_kernel(const float* __restrict__ z,
                                  const float* __restrict__ emb,
                                  float* __restrict__ out) {
    __shared__ float red[256];
    const int n = blockIdx.x * blockDim.x + threadIdx.x;   // one thread per latent
    const int b  = n >> 10;            // n / 1024
    const int hw = n & (HWSZ - 1);
    const int k  = (int)out[OUT_IDX + n];

    const float4* z4 = (const float4*)(z + n * D_DIM);
    const float4* e4 = (const float4*)(emb + k * D_DIM);
    float* dw = out + OUT_EMA + k * D_DIM;
    float lsum = 0.0f;
    #pragma unroll
    for (int q = 0; q < D_DIM / 4; ++q) {
        float4 zv = z4[q];
        float4 ev = e4[q];
        int d = q * 4;
        // out[b][d][h][w] = quantized (straight-through == embedding value)
        out[OUT_Q + (b * D_DIM + d + 0) * HWSZ + hw] = ev.x;
        out[OUT_Q + (b * D_DIM + d + 1) * HWSZ + hw] = ev.y;
        out[OUT_Q + (b * D_DIM + d + 2) * HWSZ + hw] = ev.z;
        out[OUT_Q + (b * D_DIM + d + 3) * HWSZ + hw] = ev.w;
        float dx = ev.x - zv.x, dy = ev.y - zv.y, dz = ev.z - zv.z, dwv = ev.w - zv.w;
        lsum += dx * dx + dy * dy + dz * dz + dwv * dwv;
        atomicAdd(dw + d + 0, zv.x);
        atomicAdd(dw + d + 1, zv.y);
        atomicAdd(dw + d + 2, zv.z);
        atomicAdd(dw + d + 3, zv.w);
    }
    atomicAdd(out + OUT_CS + k, 1.0f);

    // block-reduce the commitment-loss partial sum
    red[threadIdx.x] = lsum;
    __syncthreads();
    for (int s = 128; s > 0; s >>= 1) {
        if (threadIdx.x < s) red[threadIdx.x] += red[threadIdx.x + s];
        __syncthreads();
    }
    if (threadIdx.x == 0) atomicAdd(out + OUT_LOSS, red[0]);
}

// ---------------- kernel 4: new_cluster_size + global sum n ----------------
__global__ void vq_cluster_kernel(const float* __restrict__ cluster_size,
                                  float* __restrict__ out, float* __restrict__ ws) {
    __shared__ float red[256];
    const int k = blockIdx.x * blockDim.x + threadIdx.x;   // 1024 threads
    float ncs = cluster_size[k] * DECAY + out[OUT_CS + k] * ONE_MINUS;
    out[OUT_CS + k] = ncs;
    red[threadIdx.x] = ncs;
    __syncthreads();
    for (int s = 128; s > 0; s >>= 1) {
        if (threadIdx.x < s) red[threadIdx.x] += red[threadIdx.x + s];
        __syncthreads();
    }
    if (threadIdx.x == 0) atomicAdd(ws + WS_NACC, red[0]);
}

// ---------------- kernel 5: finalize ema / embeddings / loss ----------------
__global__ void vq_finalize_kernel(const float* __restrict__ ema,
                                   const float* __restrict__ ws,
                                   float* __restrict__ out) {
    const int tid = blockIdx.x * blockDim.x + threadIdx.x;  // 65536 threads
    const int k = tid >> 6;   // / 64
    const float n   = ws[WS_NACC];
    const float ncs = out[OUT_CS + k];
    const float dw  = out[OUT_EMA + tid];
    const float newema = ema[tid] * DECAY + dw * ONE_MINUS;
    out[OUT_EMA + tid] = newema;
    const float cs = (ncs + EPSV) / (n + (float)K_CODES * EPSV) * n;
    out[OUT_EMB + tid] = newema / cs;
    if (tid == 0)
        out[OUT_LOSS] = out[OUT_LOSS] * (COMMIT / (float)(N_TOT * D_DIM));
}

extern "C" void kernel_launch(void* const* d_in, const int* in_sizes, int n_in,
                              void* d_out, int out_size, void* d_ws, size_t ws_size,
                              hipStream_t stream) {
    const float* z    = (const float*)d_in[0];
    const float* emb  = (const float*)d_in[1];
    const float* ema  = (const float*)d_in[2];
    const float* csz  = (const float*)d_in[3];
    float* out = (float*)d_out;
    float* ws  = (float*)d_ws;
    (void)in_sizes; (void)n_in; (void)out_size; (void)ws_size;

    vq_init_kernel<<<256, 256, 0, stream>>>(emb, out, ws);
    // 4096 waves (16 rows each), 8 waves per 256-thread block -> 512 blocks
    vq_argmin_wmma_kernel<<<512, 256, 0, stream>>>(z, emb, ws, out);
    vq_scatter_kernel<<<N_TOT / 256, 256, 0, stream>>>(z, emb, out);
    vq_cluster_kernel<<<K_CODES / 256, 256, 0, stream>>>(csz, out, ws);
    vq_finalize_kernel<<<256, 256, 0, stream>>>(ema, ws, out);
}